// HeatEquation_45578192945898
// MI455X (gfx1250) — compile-verified
//
#include <hip/hip_runtime.h>
#include <stdint.h>

// ---------------- problem constants ----------------
#define NN 512
#define PXY 16
#define TILE 32
#define CHUNK 12                  // time steps fused per launch (= halo width)
#define REG (TILE + 2*CHUNK)      // 56: tile + halo region held in LDS
#define BT 256                    // threads per block (8 wave32)
#define TOTAL_STEPS 399           // ys[1:] = states after steps 1..399
#define KITER ((REG*REG + BT - 1) / BT)   // 13 region points per thread

// DT / DX^2 folded into the diffusivity map (computed in double, like Python)
constexpr double kDT  = 5e-07;
constexpr double kDX  = 1.0 / 511.0;
constexpr double kFac = kDT / (kDX * kDX);   // ~0.13056

// ---------------- CDNA5 feature detection (device pass only) ----------------
#if defined(__AMDGCN__)
# if __has_builtin(__builtin_amdgcn_global_load_async_to_lds_b32)
#  define HAS_ASYNC_LDS 1
# else
#  define HAS_ASYNC_LDS 0
# endif
# if __has_builtin(__builtin_amdgcn_s_wait_asynccnt)
#  define HAS_WAIT_ASYNC 1
# else
#  define HAS_WAIT_ASYNC 0
# endif
#else
# define HAS_ASYNC_LDS 0
# define HAS_WAIT_ASYNC 0
#endif

// Builtin signature (probe-confirmed): (AS1 int* src, LDS int* dst, Ii off, Ii cpol)
typedef __attribute__((address_space(3))) int* as3_i32;
typedef __attribute__((address_space(1))) int* as1_i32;
// Generic->LDS: low 32 bits of a generic LDS pointer are the LDS offset.
#define TO_AS3(p) ((as3_i32)(unsigned int)(uintptr_t)(p))
#define TO_AS1(p) ((as1_i32)(uintptr_t)(p))

#if HAS_WAIT_ASYNC
# define WAIT_ASYNC() __builtin_amdgcn_s_wait_asynccnt(0)
#else
# define WAIT_ASYNC() asm volatile("s_wait_asynccnt 0x0" ::: "memory")
#endif

// ---------------- kernel: temporally-blocked stencil chunk ----------------
// Advances `ns` (<= CHUNK) steps from `src`, writing state-after-step (step0+s)
// into out frame (step0+s-1), s = 1..ns. Each block owns a 32x32 tile and keeps
// a 56x56 halo region in LDS for the whole chunk (overlapped tiling).
__global__ __launch_bounds__(BT)
void heat_chunk_kernel(const float* __restrict__ src,
                       const float* __restrict__ cmap,
                       float* __restrict__ out,
                       int step0, int ns) {
    __shared__ float bufA[REG * REG];
    __shared__ float bufB[REG * REG];
    __shared__ float cbuf[REG * REG];

    const int tid = threadIdx.x;
    const int gx0 = (int)blockIdx.x * TILE - CHUNK;
    const int gy0 = (int)blockIdx.y * TILE - CHUNK;

    // ---- phase 1: launch async global->LDS DMA for the field region ----
    unsigned mask = 0;   // per-thread loop-invariant "compute this point" bits
    {
        int i = 0;
        for (int k = tid; k < REG * REG; k += BT, ++i) {
            int ry = k / REG, rx = k - ry * REG;
            int gy = gy0 + ry, gx = gx0 + rx;
            bool in = (gx >= 0) && (gx < NN) && (gy >= 0) && (gy < NN);
            bool compute = (ry > 0) && (ry < REG - 1) && (rx > 0) && (rx < REG - 1) &&
                           (gy > 0) && (gy < NN - 1) && (gx > 0) && (gx < NN - 1);
            mask |= ((unsigned)compute) << i;
#if HAS_ASYNC_LDS
            if (in) {
                __builtin_amdgcn_global_load_async_to_lds_b32(
                    TO_AS1(src + (size_t)gy * NN + gx),
                    TO_AS3(&bufA[k]), 0, 0);
            } else {
                bufA[k] = 0.0f;
            }
#else
            bufA[k] = in ? src[(size_t)gy * NN + gx] : 0.0f;
#endif
        }
    }

    // ---- phase 2: coefficient tile (overlaps with the async DMA) ----
    for (int k = tid; k < REG * REG; k += BT) {
        int ry = k / REG, rx = k - ry * REG;
        int gy = gy0 + ry, gx = gx0 + rx;
        int cy = gy < 0 ? 0 : (gy > NN - 1 ? NN - 1 : gy);
        int cx = gx < 0 ? 0 : (gx > NN - 1 ? NN - 1 : gx);
        cbuf[k] = cmap[cy * NN + cx];
    }

#if HAS_ASYNC_LDS
    WAIT_ASYNC();
#endif
    __syncthreads();

    float* cur = bufA;
    float* nxt = bufB;

    for (int s = 1; s <= ns; ++s) {
        // compute over the whole region; validity shrinks by 1 ring/step,
        // but halo (CHUNK) >= ns so the central TILE stays valid throughout.
        {
            int i = 0;
            for (int k = tid; k < REG * REG; k += BT, ++i) {
                float v = 0.0f;
                if ((mask >> i) & 1u) {
                    float t0 = cur[k];
                    float lap = cur[k - REG] + cur[k + REG]
                              + cur[k - 1]   + cur[k + 1] - 4.0f * t0;
                    v = fmaf(cbuf[k], lap, t0);
                }
                nxt[k] = v;
            }
        }
        __syncthreads();

        // stream this step's central tile to its output frame (non-temporal:
        // 399 MB write stream should not thrash L2)
        float* fr = out + (size_t)(step0 + s - 1) * (NN * NN);
        for (int k = tid; k < TILE * TILE; k += BT) {
            int ty = k / TILE, tx = k - ty * TILE;
            int gy = (int)blockIdx.y * TILE + ty;
            int gx = (int)blockIdx.x * TILE + tx;
            __builtin_nontemporal_store(nxt[(ty + CHUNK) * REG + (tx + CHUNK)],
                                        fr + (size_t)gy * NN + gx);
        }
        float* t = cur; cur = nxt; nxt = t;
        // single barrier per step is sufficient: next compute writes the buffer
        // nobody reads anymore, and reads the buffer the stores also read.
    }
}

// ---------------- kernel: diffusivity map (bilinear align_corners 16->512) ----------------
__global__ __launch_bounds__(256)
void heat_coef_kernel(const float* __restrict__ alpha, float* __restrict__ cmap) {
    int gx = blockIdx.x * blockDim.x + threadIdx.x;
    int gy = blockIdx.y * blockDim.y + threadIdx.y;
    if (gx >= NN || gy >= NN) return;
    const float scale = 15.0f / 511.0f;
    float yf = (float)gy * scale;
    float xf = (float)gx * scale;
    int y0 = (int)floorf(yf); if (y0 < 0) y0 = 0; if (y0 > PXY - 2) y0 = PXY - 2;
    int x0 = (int)floorf(xf); if (x0 < 0) x0 = 0; if (x0 > PXY - 2) x0 = PXY - 2;
    float wy = yf - (float)y0;
    float wx = xf - (float)x0;
    float a00 = alpha[y0 * PXY + x0];
    float a01 = alpha[y0 * PXY + x0 + 1];
    float a10 = alpha[(y0 + 1) * PXY + x0];
    float a11 = alpha[(y0 + 1) * PXY + x0 + 1];
    float d = a00 * (1.0f - wy) * (1.0f - wx) + a01 * (1.0f - wy) * wx
            + a10 * wy * (1.0f - wx)         + a11 * wy * wx;
    cmap[gy * NN + gx] = (float)kFac * d;
}

// ---------------- TDM probe (never launched; codegen-only) ----------------
typedef unsigned int u32x4 __attribute__((ext_vector_type(4)));
typedef int i32x4 __attribute__((ext_vector_type(4)));
typedef int i32x8 __attribute__((ext_vector_type(8)));

__global__ void cdna5_tdm_probe(float* o) {
#if defined(__AMDGCN__) && __has_builtin(__builtin_amdgcn_tensor_load_to_lds)
    u32x4 g0 = {0, 0, 0, 0};
    i32x8 g1 = {0, 0, 0, 0, 0, 0, 0, 0};
    i32x4 g2 = {0, 0, 0, 0};
    i32x4 g3 = {0, 0, 0, 0};
# if __clang_major__ >= 23
    i32x8 g4 = {0, 0, 0, 0, 0, 0, 0, 0};
    __builtin_amdgcn_tensor_load_to_lds(g0, g1, g2, g3, g4, 0);
# else
    __builtin_amdgcn_tensor_load_to_lds(g0, g1, g2, g3, 0);
# endif
# if __has_builtin(__builtin_amdgcn_s_wait_tensorcnt)
    __builtin_amdgcn_s_wait_tensorcnt(0);
# endif
#endif
    if (o) o[0] = 1.0f;
}

// ---------------- launch ----------------
extern "C" void kernel_launch(void* const* d_in, const int* in_sizes, int n_in,
                              void* d_out, int out_size, void* d_ws, size_t ws_size,
                              hipStream_t stream) {
    const float* u0    = (const float*)d_in[0];   // 512*512
    const float* alpha = (const float*)d_in[1];   // 16*16
    float* out  = (float*)d_out;                  // 399*512*512
    float* cmap = (float*)d_ws;                   // 512*512 scratch (1 MB)

    // 1) diffusivity map (recomputed every call; ws is not preserved)
    heat_coef_kernel<<<dim3(NN / 16, NN / 16), dim3(16, 16), 0, stream>>>(alpha, cmap);

    // 2) 399 steps in chunks of CHUNK; frame f holds state after step f+1.
    int done = 0;
    while (done < TOTAL_STEPS) {
        int ns = TOTAL_STEPS - done;
        if (ns > CHUNK) ns = CHUNK;
        const float* src = (done == 0) ? u0 : out + (size_t)(done - 1) * (NN * NN);
        heat_chunk_kernel<<<dim3(NN / TILE, NN / TILE), dim3(BT), 0, stream>>>(
            src, cmap, out, done, ns);
        done += ns;
    }
}